// NeuralConvNetwork_v2_81844896793181
// MI455X (gfx1250) — compile-verified
//
#include <hip/hip_runtime.h>

typedef float v2f __attribute__((ext_vector_type(2)));
typedef float v8f __attribute__((ext_vector_type(8)));

#define WAVES 8
#define TPB (WAVES * 32)
#define BLOCKM 32

// ---------------------------------------------------------------- utilities
__global__ void zero_kernel(float* __restrict__ p, int n) {
  int i = blockIdx.x * blockDim.x + threadIdx.x;
  if (i < n) p[i] = 0.f;
}

// Build fused weight, K-pair interleaved for single-b64 WMMA B fragments:
//   Vpk[(k/2)*(2*NPAD) + n*2 + (k&1)] = [Ws ; Wd_deg][k][n]  (zero padded)
// and fused bias[n] = bs[n] + bd_deg[n].
__global__ void build_v_kernel(const float* __restrict__ Ws,
                               const float* __restrict__ Wd,
                               const float* __restrict__ bs,
                               const float* __restrict__ bd,
                               int din, int dout, int KPAD, int NPAD,
                               float* __restrict__ Vpk, float* __restrict__ bias) {
  int idx = blockIdx.x * blockDim.x + threadIdx.x;
  int tot = KPAD * NPAD;
  if (idx < tot) {
    int k = idx / NPAD, n = idx - k * NPAD;
    float v = 0.f;
    if (n < dout) {
      if (k < din)              v = Ws[k * dout + n];
      else if (k < 2 * din + 6) v = Wd[(k - din) * dout + n];
    }
    Vpk[(size_t)(k >> 1) * (2 * NPAD) + n * 2 + (k & 1)] = v;
  }
  if (idx < NPAD) bias[idx] = (idx < dout) ? (bs[idx] + bd[idx]) : 0.f;
}

// One block per atom row: u[i] = [ x[i] | sum_j x[an[j]] | sum_j bond[bn[j]] | 0pad ]
__global__ void gather_concat_kernel(const float* __restrict__ X, int ldx, int din,
                                     const float* __restrict__ bonds,
                                     const int* __restrict__ an,
                                     const int* __restrict__ bn,
                                     int deg, int row_base,
                                     float* __restrict__ U, int KU) {
  __shared__ int aidx[4];
  __shared__ int bidx[4];
  int r = blockIdx.x;
  int t = threadIdx.x;
  if (t < deg) aidx[t] = an[r * deg + t];
  else if (t >= 32 && t < 32 + deg) bidx[t - 32] = bn[r * deg + (t - 32)];
  __syncthreads();
  int c = t;
  if (c >= KU) return;
  size_t gi = (size_t)(row_base + r);
  float s = 0.f;
  if (c < din) {
    s = X[gi * ldx + c];                                   // self features
  } else if (c < 2 * din) {
    int cc = c - din;
    for (int j = 0; j < deg; ++j) s += X[(size_t)aidx[j] * ldx + cc];
  } else if (c < 2 * din + 6) {
    int cc = c - 2 * din;
    for (int j = 0; j < deg; ++j) s += bonds[(size_t)bidx[j] * 6 + cc];
  }
  U[gi * KU + c] = s;
}

// ------------------------------------------------------------- fused GEMM
// Block = 32 output rows x NPAD cols. Each wave runs two stacked 16x16 tiles
// (shared B fragment, two independent WMMA chains). t = u @ V + bias ;
// row L2-normalize ; relu ; store (or run-length molecule segment-sum).
template <int KPAD, int NPAD, bool FINAL>
__global__ __launch_bounds__(TPB)
void gemm_norm_kernel(const float* __restrict__ U,
                      const float* __restrict__ Vpk,
                      const float* __restrict__ bias,
                      int row_base, int n_rows,
                      float* __restrict__ Xout,
                      const int* __restrict__ mol_ids,
                      float* __restrict__ Out) {
  constexpr int NT = NPAD / 16;
  constexpr int AS = KPAD + 2;           // even: keeps (k,k+1) pair 8B aligned
  __shared__ float As[BLOCKM][AS];
  __shared__ float Ts[BLOCKM][NPAD + 4];
  __shared__ float rowscale[BLOCKM];
  __shared__ int   mol_s[BLOCKM];

  const int tid  = threadIdx.x;
  const int wave = tid >> 5;
  const int lane = tid & 31;
  const int m    = lane & 15;            // A row / D col within tile
  const int kh   = lane >> 4;            // K half (f32 WMMA layout)
  const int blockRow = blockIdx.x * BLOCKM;

  // Stage BLOCKM x KPAD A-tile (u rows) into LDS, zero-fill guard rows.
  for (int idx = tid; idx < BLOCKM * KPAD; idx += TPB) {
    int r = idx / KPAD, c = idx - r * KPAD;
    int gr = blockRow + r;
    As[r][c] = (gr < n_rows) ? U[(size_t)(row_base + gr) * KPAD + c] : 0.f;
  }
  if (FINAL && tid < BLOCKM) {
    int gr = blockRow + tid;
    mol_s[tid] = (gr < n_rows) ? mol_ids[row_base + gr] : -1;
  }
  __syncthreads();

  // Each wave sweeps 16-col N-tiles; per tile two M-stacked WMMA chains.
  for (int t = wave; t < NT; t += WAVES) {
    v8f acc0 = {}, acc1 = {};
    const float* vb = Vpk + (size_t)(t * 16 + m) * 2;
#pragma unroll
    for (int k0 = 0; k0 < KPAD; k0 += 4) {
      int kc = k0 + 2 * kh;              // A/B: VGPR0=K{0|2}, VGPR1=K{1|3}
      v2f a0 = *(const v2f*)&As[m][kc];
      v2f a1 = *(const v2f*)&As[m + 16][kc];
      v2f b  = *(const v2f*)(vb + (size_t)((k0 >> 1) + kh) * (2 * NPAD));
      acc0 = __builtin_amdgcn_wmma_f32_16x16x4_f32(
          false, a0, false, b, (short)0, acc0, false, false);
      acc1 = __builtin_amdgcn_wmma_f32_16x16x4_f32(
          false, a1, false, b, (short)0, acc1, false, false);
    }
    float bb = bias[t * 16 + m];
#pragma unroll
    for (int j = 0; j < 8; ++j) {        // D: vgpr j -> row j + 8*kh
      Ts[j + 8 * kh][t * 16 + m]      = acc0[j] + bb;
      Ts[16 + j + 8 * kh][t * 16 + m] = acc1[j] + bb;
    }
  }
  __syncthreads();

  // Row-wise L2 norm: 8 threads per row, shfl_xor tree stays in-wave.
  {
    int g = tid >> 3, li = tid & 7;
    float s = 0.f;
    for (int c = li; c < NPAD; c += 8) { float v = Ts[g][c]; s += v * v; }
    s += __shfl_xor(s, 1);
    s += __shfl_xor(s, 2);
    s += __shfl_xor(s, 4);
    if (li == 0) rowscale[g] = 1.0f / fmaxf(sqrtf(s), 1e-12f);
  }
  __syncthreads();

  if (!FINAL) {
    // relu(t * scale) -> next-layer activations
    for (int idx = tid; idx < BLOCKM * NPAD; idx += TPB) {
      int r = idx / NPAD, c = idx - r * NPAD;
      int gr = blockRow + r;
      if (gr < n_rows)
        Xout[(size_t)(row_base + gr) * NPAD + c] = fmaxf(Ts[r][c] * rowscale[r], 0.f);
    }
  } else {
    // Molecule segment-sum: atoms are consecutive per molecule, so a 32-row
    // block spans <=3 molecules -> run-length reduce in LDS, few atomics.
    for (int c = tid; c < NPAD; c += TPB) {
      float accv = 0.f;
      int cur = -1;
#pragma unroll 4
      for (int r = 0; r < BLOCKM; ++r) {
        int mid = mol_s[r];
        if (mid < 0) break;              // trailing guard rows
        float v = fmaxf(Ts[r][c] * rowscale[r], 0.f);
        if (mid != cur) {
          if (cur >= 0) atomicAdd(Out + (size_t)cur * NPAD + c, accv);
          cur = mid;
          accv = 0.f;
        }
        accv += v;
      }
      if (cur >= 0) atomicAdd(Out + (size_t)cur * NPAD + c, accv);
    }
  }
}

// ------------------------------------------------------------------ driver
extern "C" void kernel_launch(void* const* d_in, const int* in_sizes, int n_in,
                              void* d_out, int out_size, void* d_ws, size_t ws_size,
                              hipStream_t stream) {
  const float* atom = (const float*)d_in[0];
  const float* bond = (const float*)d_in[1];
  const int* an[4]  = {(const int*)d_in[2], (const int*)d_in[4],
                       (const int*)d_in[6], (const int*)d_in[8]};
  const int* bnn[4] = {(const int*)d_in[3], (const int*)d_in[5],
                       (const int*)d_in[7], (const int*)d_in[9]};
  const int* mol_ids = (const int*)d_in[10];
  const float* Ws[3] = {(const float*)d_in[11], (const float*)d_in[15], (const float*)d_in[19]};
  const float* bs[3] = {(const float*)d_in[12], (const float*)d_in[16], (const float*)d_in[20]};
  const float* Wd[3] = {(const float*)d_in[13], (const float*)d_in[17], (const float*)d_in[21]};
  const float* bd[3] = {(const float*)d_in[14], (const float*)d_in[18], (const float*)d_in[22]};

  const int deg_count[4] = {100000, 175000, 150000, 75000};
  const int deg_base[4]  = {0, 100000, 275000, 425000};
  const int NA = 500000;

  const int din_l[3]  = {62, 100, 100};
  const int KPAD_l[3] = {132, 208, 208};   // round_up(2*din+6, 4)
  const int NPAD_l[3] = {112, 112, 512};   // round_up(dout, 16)
  const int dout_l[3] = {100, 100, 512};

  // workspace layout (floats)
  float* ws   = (float*)d_ws;
  float* Ubuf = ws;                          // 500000 * 208
  float* Xbuf = Ubuf + (size_t)NA * 208;     // 500000 * 112  (layer activations)
  float* Vbuf = Xbuf + (size_t)NA * 112;     // fused packed weights + biases (~2.3 MB)
  float* Vp[3][4];
  float* Bp[3][4];
  {
    float* p = Vbuf;
    for (int l = 0; l < 3; ++l)
      for (int d = 0; d < 4; ++d) { Vp[l][d] = p; p += (size_t)KPAD_l[l] * NPAD_l[l]; }
    for (int l = 0; l < 3; ++l)
      for (int d = 0; d < 4; ++d) { Bp[l][d] = p; p += 512; }
  }

  // 1) build fused padded packed weights (L2-resident, reused by all blocks)
  for (int l = 0; l < 3; ++l) {
    int din = din_l[l], dout = dout_l[l], KP = KPAD_l[l], NP = NPAD_l[l];
    for (int d = 0; d < 4; ++d) {
      int tot = KP * NP;
      build_v_kernel<<<(tot + 255) / 256, 256, 0, stream>>>(
          Ws[l], Wd[l] + (size_t)d * (din + 6) * dout,
          bs[l], bd[l] + (size_t)d * dout,
          din, dout, KP, NP, Vp[l][d], Bp[l][d]);
    }
  }

  // 2) zero output (final layer accumulates with atomics)
  zero_kernel<<<(out_size + 255) / 256, 256, 0, stream>>>((float*)d_out, out_size);

  // 3) layers: gather/concat -> fused WMMA GEMM + norm + relu (+ segment sum)
  for (int l = 0; l < 3; ++l) {
    const float* Xsrc = (l == 0) ? atom : Xbuf;
    int ldx = (l == 0) ? 62 : 112;
    int din = din_l[l], KP = KPAD_l[l];

    for (int d = 0; d < 4; ++d)
      gather_concat_kernel<<<deg_count[d], 256, 0, stream>>>(
          Xsrc, ldx, din, bond, an[d], bnn[d], d + 1, deg_base[d], Ubuf, KP);

    for (int d = 0; d < 4; ++d) {
      int nblk = (deg_count[d] + BLOCKM - 1) / BLOCKM;
      if (l == 0)
        gemm_norm_kernel<132, 112, false><<<nblk, TPB, 0, stream>>>(
            Ubuf, Vp[l][d], Bp[l][d], deg_base[d], deg_count[d], Xbuf, mol_ids, (float*)d_out);
      else if (l == 1)
        gemm_norm_kernel<208, 112, false><<<nblk, TPB, 0, stream>>>(
            Ubuf, Vp[l][d], Bp[l][d], deg_base[d], deg_count[d], Xbuf, mol_ids, (float*)d_out);
      else
        gemm_norm_kernel<208, 512, true><<<nblk, TPB, 0, stream>>>(
            Ubuf, Vp[l][d], Bp[l][d], deg_base[d], deg_count[d], Xbuf, mol_ids, (float*)d_out);
    }
  }
}